// Gcn_81990925681140
// MI455X (gfx1250) — compile-verified
//
#include <hip/hip_runtime.h>

// ---------------------------------------------------------------------------
// 3-layer GCN for MI455X (gfx1250, wave32).
//   per layer: H = act(X) @ W  (dense, WMMA f32 16x16x4)
//              AGG = b  ;  AGG[dst[e]] += vals[e] * H[src[e]]  (L2 atomics)
//              next X = leaky(AGG)   (fused into next GEMM's load)
// Sizes: N=100000 nodes, E=3.2M edges, feat 512 -> 256 -> 256 -> 40.
// ---------------------------------------------------------------------------

typedef __attribute__((ext_vector_type(2))) float v2f;
typedef __attribute__((ext_vector_type(8))) float v8f;

#define N_NODES   100000
#define N_EDGES   3200000
#define NEG_SLOPE 0.01f

__device__ __forceinline__ float leaky(float x) {
  return x > 0.f ? x : NEG_SLOPE * x;
}

// ---------------- Dense GEMM via V_WMMA_F32_16X16X4_F32 --------------------
// H[N,M] = act(X[N,K]) @ W[K,M], all row-major fp32. K is a multiple of 32.
// Block: 256 threads = 8 waves. Tile: 128 rows x 16 cols; each wave owns a
// 16x16 tile and loops K in steps of 4 (8 WMMA per 32-wide K stage).
#define KT   32
#define XPAD 2   // row stride 34 floats: distinct banks for 16-row column
                 // reads, and 136B rows keep even-k pairs 8B aligned (b64).

__global__ __launch_bounds__(256) void gemm_wmma_f32(
    const float* __restrict__ X, const float* __restrict__ W,
    float* __restrict__ H, int N, int K, int M, int leaky_in)
{
  __shared__ float Xs[128][KT + XPAD];
  __shared__ float Ws[KT][16];

  const int tid     = threadIdx.x;
  const int rowBase = blockIdx.y * 128;
  const int colBase = blockIdx.x * 16;

  const int wave = tid >> 5;       // wave32 on gfx1250
  const int lane = tid & 31;
  const int half = lane >> 4;      // 0: K0/K1 pair, 1: K2/K3 pair (ISA layout)
  const int lid  = lane & 15;
  const int wrow = wave * 16;

  v8f acc = {0.f, 0.f, 0.f, 0.f, 0.f, 0.f, 0.f, 0.f};

  for (int k0 = 0; k0 < K; k0 += KT) {
    // Stage X tile (128 x 32) with coalesced float4 loads; fuse previous
    // layer's leaky-ReLU here so activations never need a separate pass.
    for (int i = tid; i < 128 * (KT / 4); i += 256) {
      int r  = i >> 3;             // 8 float4 per row
      int c4 = (i & 7) << 2;
      int gr = rowBase + r;
      float4 v = make_float4(0.f, 0.f, 0.f, 0.f);
      if (gr < N) v = *(const float4*)(X + (size_t)gr * K + k0 + c4);
      if (leaky_in) {
        v.x = leaky(v.x); v.y = leaky(v.y); v.z = leaky(v.z); v.w = leaky(v.w);
      }
      Xs[r][c4 + 0] = v.x; Xs[r][c4 + 1] = v.y;
      Xs[r][c4 + 2] = v.z; Xs[r][c4 + 3] = v.w;
    }
    // Stage W panel (32 x 16), column-guarded (M=40 tail tile).
    for (int i = tid; i < KT * 4; i += 256) {
      int r  = i >> 2;
      int c4 = (i & 3) << 2;
      const float* wr = W + (size_t)(k0 + r) * M;
#pragma unroll
      for (int j = 0; j < 4; ++j) {
        int gc = colBase + c4 + j;
        Ws[r][c4 + j] = (gc < M) ? wr[gc] : 0.f;
      }
    }
    __syncthreads();

#pragma unroll
    for (int kk = 0; kk < KT; kk += 4) {
      // A 16x4 f32: lanes 0-15 hold (row=lid, K=kk..kk+1), lanes 16-31 hold
      // (row=lid, K=kk+2..kk+3).  B 4x16 mirrors with col=lid.
      int ka = kk + 2 * half;
      v2f a, b;
      a.x = Xs[wrow + lid][ka];  a.y = Xs[wrow + lid][ka + 1];
      b.x = Ws[ka][lid];         b.y = Ws[ka + 1][lid];
      acc = __builtin_amdgcn_wmma_f32_16x16x4_f32(
          /*neg_a=*/false, a, /*neg_b=*/false, b,
          /*c_mod=*/(short)0, acc, /*reuse_a=*/false, /*reuse_b=*/false);
    }
    __syncthreads();
  }

  // D layout: VGPR j = row (j + 8*half) of the 16x16 tile, col = lid.
  int col = colBase + lid;
  if (col < M) {
#pragma unroll
    for (int j = 0; j < 8; ++j) {
      int row = rowBase + wrow + half * 8 + j;
      if (row < N) H[(size_t)row * M + col] = acc[j];
    }
  }
}

// ---------------- AGG initialization: AGG[i, d] = b[d] ---------------------
__global__ void init_bias(float* __restrict__ AGG, const float* __restrict__ b,
                          long total, int D)
{
  long stride = (long)gridDim.x * blockDim.x;
  for (long i = (long)blockIdx.x * blockDim.x + threadIdx.x; i < total; i += stride)
    AGG[i] = b[i % D];
}

// ---------------- SpMM scatter: AGG[dst[e]] += vals[e] * H[src[e]] ---------
// One edge per wave32: lanes stride the row -> coalesced gather, then
// per-element global_atomic_add_f32 (no return) into L2. The 102MB row matrix
// fits MI455X's 192MB L2, so both sides of this stay on-chip.
__global__ __launch_bounds__(256) void spmm_scatter(
    const int* __restrict__ src, const int* __restrict__ dst,
    const float* __restrict__ vals, const float* __restrict__ H,
    float* __restrict__ AGG, int E, int D)
{
  int lane = threadIdx.x & 31;
  int wid  = (blockIdx.x * blockDim.x + threadIdx.x) >> 5;
  int nw   = (gridDim.x * blockDim.x) >> 5;
  for (int e = wid; e < E; e += nw) {
    int   s = src[e];
    int   d = dst[e];
    float v = vals[e];
    const float* hrow = H   + (size_t)s * D;
    float*       arow = AGG + (size_t)d * D;
    for (int c = lane; c < D; c += 32)
      atomicAdd(&arow[c], v * hrow[c]);
  }
}

// ---------------- Final activation on d_out --------------------------------
__global__ void leaky_inplace(float* __restrict__ x, long total)
{
  long stride = (long)gridDim.x * blockDim.x;
  for (long i = (long)blockIdx.x * blockDim.x + threadIdx.x; i < total; i += stride)
    x[i] = leaky(x[i]);
}

// ---------------------------------------------------------------------------
extern "C" void kernel_launch(void* const* d_in, const int* in_sizes, int n_in,
                              void* d_out, int out_size, void* d_ws, size_t ws_size,
                              hipStream_t stream)
{
  (void)in_sizes; (void)n_in; (void)out_size; (void)ws_size;

  const float* feat = (const float*)d_in[0];
  const int*   esrc = (const int*)  d_in[1];
  const int*   edst = (const int*)  d_in[2];
  const float* eval = (const float*)d_in[3];
  const float* W1   = (const float*)d_in[4];
  const float* b1   = (const float*)d_in[5];
  const float* Wh   = (const float*)d_in[6];
  const float* bh   = (const float*)d_in[7];
  const float* W2   = (const float*)d_in[8];
  const float* b2   = (const float*)d_in[9];
  float*       out  = (float*)d_out;

  const int N = N_NODES, E = N_EDGES;
  const int F = 512, HID = 256, C = 40;

  float* bufA = (float*)d_ws;                  // GEMM outputs   [N, 256]
  float* bufB = bufA + (size_t)N * HID;        // AGG / next X   [N, 256]

  dim3 blk(256);
  dim3 gGemmH((HID + 15) / 16, (N + 127) / 128);   // 16 x 782
  dim3 gGemmC((C   + 15) / 16, (N + 127) / 128);   //  3 x 782
  dim3 gSpmm(8192);                                 // 65536 waves grid-stride
  dim3 gElem(4096);

  // ---- Layer 1: h = leaky(A @ (X @ W1) + b1) ----
  gemm_wmma_f32<<<gGemmH, blk, 0, stream>>>(feat, W1, bufA, N, F, HID, 0);
  init_bias    <<<gElem,  blk, 0, stream>>>(bufB, b1, (long)N * HID, HID);
  spmm_scatter <<<gSpmm,  blk, 0, stream>>>(esrc, edst, eval, bufA, bufB, E, HID);
  // leaky fused into next GEMM's X load

  // ---- Layer 2 ----
  gemm_wmma_f32<<<gGemmH, blk, 0, stream>>>(bufB, Wh, bufA, N, HID, HID, 1);
  init_bias    <<<gElem,  blk, 0, stream>>>(bufB, bh, (long)N * HID, HID);
  spmm_scatter <<<gSpmm,  blk, 0, stream>>>(esrc, edst, eval, bufA, bufB, E, HID);

  // ---- Layer 3 (scatter straight into d_out) ----
  gemm_wmma_f32<<<gGemmC, blk, 0, stream>>>(bufB, W2, bufA, N, HID, C, 1);
  init_bias    <<<gElem,  blk, 0, stream>>>(out, b2, (long)N * C, C);
  spmm_scatter <<<gSpmm,  blk, 0, stream>>>(esrc, edst, eval, bufA, out, E, C);
  leaky_inplace<<<gElem,  blk, 0, stream>>>(out, (long)N * C);
}